// FAGCN_79044578116363
// MI455X (gfx1250) — compile-verified
//
#include <hip/hip_runtime.h>
#include <math.h>

#define NNODE 50000
#define NEDGE 800000
#define INC 512
#define HID 64
#define OUTC 40
#define OUTP 48          // padded to 3 WMMA N-tiles
#define NLAYER 2
#define EPSV 0.3f
#define MTILES 3125      // 50000 / 16 exactly
#define KS 64            // K-slab staged in LDS for GEMM1
#define NPAIR (KS / 2)   // 32 K-pairs per slab

typedef float v2f __attribute__((ext_vector_type(2)));
typedef float v8f __attribute__((ext_vector_type(8)));

#define WMMA_F32(a, b, c) \
    __builtin_amdgcn_wmma_f32_16x16x4_f32(false, (a), false, (b), (short)0, (c), false, false)

// swizzled LDS index: pair p (0..31), col (0..63); col^p stays in 0..63
#define BIDX(p, c) ((p) * HID + (((c) ^ (p))))

__device__ __forceinline__ void atomAddF(float* p, float v) {
    __hip_atomic_fetch_add(p, v, __ATOMIC_RELAXED, __HIP_MEMORY_SCOPE_AGENT);
}

// ---------- degree / norm ----------
__global__ void k_deg(const int* __restrict__ dst, float* __restrict__ deg) {
    int e = blockIdx.x * blockDim.x + threadIdx.x;
    if (e < NEDGE) atomAddF(&deg[dst[e]], 1.0f);
}

__global__ void k_dinv(float* __restrict__ deg) {
    int n = blockIdx.x * blockDim.x + threadIdx.x;
    if (n < NNODE) {
        float d = deg[n];
        deg[n] = (d > 0.0f) ? rsqrtf(d) : 0.0f;
    }
}

__global__ void k_norm(const int* __restrict__ src, const int* __restrict__ dst,
                       const float* __restrict__ dinv, float* __restrict__ norm) {
    int e = blockIdx.x * blockDim.x + threadIdx.x;
    if (e < NEDGE) norm[e] = dinv[src[e]] * dinv[dst[e]];
}

// ---------- GEMM1: h = relu(x @ W1^T + b1) ----------
// One wave per 16-node tile computes the full 16x64 output (4 accumulators),
// so x is streamed from HBM exactly once. W1 is staged through LDS in
// 64-wide K-slabs stored as float2 K-pairs with an XOR column swizzle, so
// each B fragment is a single aligned ds_load_b64 (no repacking movs).
__global__ void __launch_bounds__(256) k_gemm1(const float* __restrict__ x,
                                               const float* __restrict__ W1,
                                               const float* __restrict__ b1,
                                               float* __restrict__ h) {
    __shared__ v2f Blds[NPAIR * HID];         // 16 KB
    int wv = threadIdx.x >> 5;
    int tm = blockIdx.x * 8 + wv;
    if (tm >= MTILES) tm = MTILES - 1;        // tail waves duplicate tile (benign)
    int lane = threadIdx.x & 31;
    int r  = lane & 15;
    int kh = (lane >> 4) << 1;                // K sub-offset {0,2} per lane half
    int ph = kh >> 1;                         // pair sub-offset {0,1}

    const float* xrow = x + (size_t)(tm * 16 + r) * INC + kh;

    v8f a0 = {0.f,0.f,0.f,0.f,0.f,0.f,0.f,0.f};
    v8f a1 = a0, a2 = a0, a3 = a0;

    for (int ks = 0; ks < INC; ks += KS) {
        __syncthreads();
        // cooperative fill: W1[col][ks+2p .. ks+2p+1] -> Blds[BIDX(p,col)]
#pragma unroll
        for (int i = 0; i < (NPAIR * HID) / 256; ++i) {   // 8 iters
            int idx = i * 256 + threadIdx.x;
            int col = idx >> 5;               // 0..63
            int p   = idx & 31;               // 0..31 (coalesced along k)
            Blds[BIDX(p, col)] = *(const v2f*)(W1 + (size_t)col * INC + ks + 2 * p);
        }
        __syncthreads();

#pragma unroll 4
        for (int k0 = 0; k0 < KS; k0 += 4) {
            v2f a = *(const v2f*)(xrow + ks + k0);
            int p = (k0 >> 1) + ph;           // pair holding (k0+kh, k0+kh+1)
            v2f b0 = Blds[BIDX(p, r +  0)];
            v2f b1v= Blds[BIDX(p, r + 16)];
            v2f b2v= Blds[BIDX(p, r + 32)];
            v2f b3v= Blds[BIDX(p, r + 48)];
            a0 = WMMA_F32(a, b0,  a0);
            a1 = WMMA_F32(a, b1v, a1);
            a2 = WMMA_F32(a, b2v, a2);
            a3 = WMMA_F32(a, b3v, a3);
        }
    }

    int mbase = tm * 16 + ((lane >> 4) << 3);
#pragma unroll
    for (int tn = 0; tn < 4; ++tn) {
        v8f acc = (tn == 0) ? a0 : (tn == 1) ? a1 : (tn == 2) ? a2 : a3;
        int col = tn * 16 + r;
        float bias = b1[col];
#pragma unroll
        for (int v = 0; v < 8; ++v) {
            float val = acc[v] + bias;
            h[(size_t)(mbase + v) * HID + col] = val > 0.0f ? val : 0.0f;
        }
    }
}

// ---------- attention scores ----------
__global__ void k_att(const float* __restrict__ h, const float* __restrict__ attl,
                      const float* __restrict__ attr, float* __restrict__ al,
                      float* __restrict__ ar) {
    int n = blockIdx.x * blockDim.x + threadIdx.x;
    if (n >= NNODE) return;
    const float* hr = h + (size_t)n * HID;
    float sl = 0.f, sr = 0.f;
#pragma unroll
    for (int k = 0; k < HID; ++k) {
        float hv = hr[k];
        sl += hv * attl[k];
        sr += hv * attr[k];
    }
    al[n] = sl;
    ar[n] = sr;
}

// ---------- per-edge weight: w = tanh(al[src]+ar[dst]) * norm ----------
__global__ void k_alpha(const int* __restrict__ src, const int* __restrict__ dst,
                        const float* __restrict__ al, const float* __restrict__ ar,
                        const float* __restrict__ norm, float* __restrict__ w) {
    int e = blockIdx.x * blockDim.x + threadIdx.x;
    if (e < NEDGE) w[e] = tanhf(al[src[e]] + ar[dst[e]]) * norm[e];
}

// ---------- edge scatter: one wave per edge, 2 channels per lane ----------
__global__ void k_edge(const int* __restrict__ src, const int* __restrict__ dst,
                       const float* __restrict__ w, const float* __restrict__ h,
                       float* __restrict__ agg) {
    int wid = (blockIdx.x * blockDim.x + threadIdx.x) >> 5;
    if (wid >= NEDGE) return;
    int lane = threadIdx.x & 31;
    int s = src[wid], d = dst[wid];
    float wv = w[wid];
    const float* hs = h + (size_t)s * HID;
    float* ad = agg + (size_t)d * HID;
    atomAddF(&ad[lane],      hs[lane]      * wv);
    atomAddF(&ad[lane + 32], hs[lane + 32] * wv);
}

// ---------- residual update: h = agg + 2*eps*raw ----------
__global__ void k_update(const float* __restrict__ agg, const float* __restrict__ raw,
                         float* __restrict__ h) {
    int i = blockIdx.x * blockDim.x + threadIdx.x;
    if (i < NNODE * HID) h[i] = agg[i] + (2.0f * EPSV) * raw[i];
}

// ---------- pad W2/b2 from 40 -> 48 rows ----------
__global__ void k_padw2(const float* __restrict__ W2, const float* __restrict__ b2,
                        float* __restrict__ W2p, float* __restrict__ b2p) {
    int i = blockIdx.x * blockDim.x + threadIdx.x;
    if (i < OUTP * HID) {
        int j = i / HID, k = i - j * HID;
        W2p[i] = (j < OUTC) ? W2[j * HID + k] : 0.0f;
        if (k == 0) b2p[j] = (j < OUTC) ? b2[j] : 0.0f;
    }
}

// ---------- GEMM2: logits[N,48] = h @ W2p^T + b2p ----------
// One wave per node tile computes all 3 N-tiles (A-fragment reuse);
// W2p is 12 KB and stays cache-resident.
__global__ void k_gemm2(const float* __restrict__ h, const float* __restrict__ W2p,
                        const float* __restrict__ b2p, float* __restrict__ logits) {
    int wid = (blockIdx.x * blockDim.x + threadIdx.x) >> 5;
    if (wid >= MTILES) return;
    int lane = threadIdx.x & 31;
    int r  = lane & 15;
    int kh = (lane >> 4) << 1;

    const float* hrow = h   + (size_t)(wid * 16 + r) * HID + kh;
    const float* w0   = W2p + (size_t)(r)      * HID + kh;
    const float* w1   = W2p + (size_t)(16 + r) * HID + kh;
    const float* w2   = W2p + (size_t)(32 + r) * HID + kh;

    v8f a0 = {0.f,0.f,0.f,0.f,0.f,0.f,0.f,0.f};
    v8f a1 = a0, a2 = a0;
#pragma unroll
    for (int k0 = 0; k0 < HID; k0 += 4) {
        v2f a = *(const v2f*)(hrow + k0);
        a0 = WMMA_F32(a, *(const v2f*)(w0 + k0), a0);
        a1 = WMMA_F32(a, *(const v2f*)(w1 + k0), a1);
        a2 = WMMA_F32(a, *(const v2f*)(w2 + k0), a2);
    }

    int mbase = wid * 16 + ((lane >> 4) << 3);
#pragma unroll
    for (int tn = 0; tn < 3; ++tn) {
        v8f acc = (tn == 0) ? a0 : (tn == 1) ? a1 : a2;
        int col = tn * 16 + r;
        float bias = b2p[col];
#pragma unroll
        for (int v = 0; v < 8; ++v)
            logits[(size_t)(mbase + v) * OUTP + col] = acc[v] + bias;
    }
}

// ---------- log_softmax over the 40 real columns ----------
__global__ void k_lsm(const float* __restrict__ logits, float* __restrict__ out) {
    int n = blockIdx.x * blockDim.x + threadIdx.x;
    if (n >= NNODE) return;
    const float* lr = logits + (size_t)n * OUTP;
    float l[OUTC];
    float m = -INFINITY;
#pragma unroll
    for (int j = 0; j < OUTC; ++j) {
        l[j] = lr[j];
        m = fmaxf(m, l[j]);
    }
    float s = 0.0f;
#pragma unroll
    for (int j = 0; j < OUTC; ++j) s += expf(l[j] - m);
    float lse = m + logf(s);
#pragma unroll
    for (int j = 0; j < OUTC; ++j) out[(size_t)n * OUTC + j] = l[j] - lse;
}

extern "C" void kernel_launch(void* const* d_in, const int* in_sizes, int n_in,
                              void* d_out, int out_size, void* d_ws, size_t ws_size,
                              hipStream_t stream) {
    const float* x    = (const float*)d_in[0];
    const int*   ei   = (const int*)d_in[1];
    const float* W1   = (const float*)d_in[2];
    const float* b1   = (const float*)d_in[3];
    const float* W2   = (const float*)d_in[4];
    const float* b2   = (const float*)d_in[5];
    const float* attL = (const float*)d_in[6];
    const float* attR = (const float*)d_in[7];
    float* out = (float*)d_out;

    const int* src = ei;            // edge_index[0]
    const int* dst = ei + NEDGE;    // edge_index[1]

    float* ws = (float*)d_ws;
    size_t off = 0;
    float* dinv   = ws + off; off += NNODE;
    float* norm   = ws + off; off += NEDGE;
    float* wbuf   = ws + off; off += NEDGE;
    float* raw    = ws + off; off += (size_t)NNODE * HID;
    float* hcur   = ws + off; off += (size_t)NNODE * HID;
    float* agg    = ws + off; off += (size_t)NNODE * HID;
    float* al     = ws + off; off += NNODE;
    float* ar     = ws + off; off += NNODE;
    float* W2p    = ws + off; off += OUTP * HID;
    float* b2p    = ws + off; off += OUTP;
    float* logits = ws + off; off += (size_t)NNODE * OUTP;

    // degree -> dinv -> per-edge norm
    hipMemsetAsync(dinv, 0, NNODE * sizeof(float), stream);
    k_deg <<<(NEDGE + 255) / 256, 256, 0, stream>>>(dst, dinv);
    k_dinv<<<(NNODE + 255) / 256, 256, 0, stream>>>(dinv);
    k_norm<<<(NEDGE + 255) / 256, 256, 0, stream>>>(src, dst, dinv, norm);

    // h0 = relu(x @ W1^T + b1)   [WMMA f32, LDS-staged W1]
    k_gemm1<<<(MTILES + 7) / 8, 256, 0, stream>>>(x, W1, b1, raw);

    const float* hin = raw;
    for (int l = 0; l < NLAYER; ++l) {
        hipMemsetAsync(agg, 0, (size_t)NNODE * HID * sizeof(float), stream);
        k_att  <<<(NNODE + 255) / 256, 256, 0, stream>>>(hin, attL + l * HID,
                                                         attR + l * HID, al, ar);
        k_alpha<<<(NEDGE + 255) / 256, 256, 0, stream>>>(src, dst, al, ar, norm, wbuf);
        k_edge <<<NEDGE / 8, 256, 0, stream>>>(src, dst, wbuf, hin, agg);
        k_update<<<(NNODE * HID + 255) / 256, 256, 0, stream>>>(agg, raw, hcur);
        hin = hcur;
    }

    // logits = h @ W2^T + b2 (padded to 48 cols), then log_softmax
    k_padw2<<<(OUTP * HID + 255) / 256, 256, 0, stream>>>(W2, b2, W2p, b2p);
    k_gemm2<<<(MTILES + 7) / 8, 256, 0, stream>>>(hcur, W2p, b2p, logits);
    k_lsm  <<<(NNODE + 255) / 256, 256, 0, stream>>>(logits, out);
}